// DeepseekV2MoE_50835232916124
// MI455X (gfx1250) — compile-verified
//
#include <hip/hip_runtime.h>
#include <math.h>
#include <stdint.h>

// DeepSeek-V2-Lite MoE dims (match reference)
#define NEXP 64
#define HDIM 2048
#define IDIM 1408
#define TOPK 6
#define TTOK 1024
#define CAP  256

typedef float v2f __attribute__((ext_vector_type(2)));
typedef float v8f __attribute__((ext_vector_type(8)));

__device__ __forceinline__ v8f wmma4(v2f a, v2f b, v8f c) {
  // D = A(16x4,f32) * B(4x16,f32) + C(16x16,f32)
  return __builtin_amdgcn_wmma_f32_16x16x4_f32(false, a, false, b, (short)0, c,
                                               false, false);
}

// Async global->LDS copy (CDNA5, ASYNCcnt-tracked). lds_off = byte offset of
// destination within the workgroup LDS allocation (low 32 bits of the flat
// LDS address per the aperture mapping).
__device__ __forceinline__ void async_load_b128(uint32_t lds_off,
                                                const float* gaddr) {
  asm volatile("global_load_async_to_lds_b128 %0, %1, off"
               :
               : "v"(lds_off), "v"(gaddr)
               : "memory");
}
__device__ __forceinline__ void wait_asynccnt0() {
  asm volatile("s_wait_asynccnt 0x0" ::: "memory");
}

__device__ __forceinline__ float silu(float v) {
  return v / (1.f + __expf(-v));
}

__global__ void init_counters(int* counters) {
  counters[threadIdx.x] = 0;
}

// One block per token; 64 threads, one per expert logit.
__global__ __launch_bounds__(64)
void router_kernel(const float* __restrict__ x, const float* __restrict__ gw,
                   int* __restrict__ counters, int* __restrict__ slot_token,
                   int* __restrict__ pair_slot, float* __restrict__ pair_weight) {
  const int t = blockIdx.x;
  const int e = threadIdx.x;
  __shared__ float lg[NEXP];

  const float* xr = x + (size_t)t * HDIM;
  float acc = 0.f;
  for (int h = 0; h < HDIM; ++h)
    acc = fmaf(xr[h], gw[(size_t)h * NEXP + e], acc);  // coalesced over e
  lg[e] = acc;
  __syncthreads();

  if (e == 0) {
    float mx = lg[0];
    for (int i = 1; i < NEXP; ++i) mx = fmaxf(mx, lg[i]);
    float p[NEXP];
    float s = 0.f;
    for (int i = 0; i < NEXP; ++i) { p[i] = __expf(lg[i] - mx); s += p[i]; }
    const float inv = 1.f / s;
    for (int i = 0; i < NEXP; ++i) p[i] *= inv;

    int   idx[TOPK];
    float w[TOPK];
    float ws = 0.f;
    for (int k = 0; k < TOPK; ++k) {           // strict > matches top_k tie rule
      int best = 0; float bv = -1.f;
      for (int i = 0; i < NEXP; ++i)
        if (p[i] > bv) { bv = p[i]; best = i; }
      idx[k] = best; w[k] = bv; ws += bv; p[best] = -2.f;
    }
    const float rinv = 1.f / ws;
    for (int k = 0; k < TOPK; ++k) {
      const int ex  = idx[k];
      const int pos = atomicAdd(&counters[ex], 1);
      const int pi  = t * TOPK + k;
      if (pos < CAP) {
        slot_token[ex * CAP + pos] = t;
        pair_slot[pi]   = ex * CAP + pos;
        pair_weight[pi] = w[k] * rinv;
      } else {
        pair_slot[pi]   = -1;
        pair_weight[pi] = 0.f;
      }
    }
  }
}

// Grouped expert GEMM. Block = 128 threads = 4 waves; block tile M=32, N=128.
// Each wave owns four 16x16 f32 accumulators (2 m-subtiles x 2 n-subtiles) so
// every B fragment is reused by two WMMAs. A tile (32x32) is double-buffered
// in LDS and filled with GLOBAL_LOAD_ASYNC_TO_LDS_B128, pipelined one K-chunk
// ahead (ASYNCcnt + one barrier per chunk).
template <int KD, int NTOT, bool GATHER, bool DO_SILU>
__global__ __launch_bounds__(128)
void expert_gemm(const float* __restrict__ Abase,
                 const int* __restrict__ slot_token,
                 const float* __restrict__ Bbase,   // [E, KD, NTOT]
                 const int* __restrict__ counters,
                 float* __restrict__ Obuf) {        // [E*CAP, NTOT]
  const int e   = blockIdx.z;
  const int cnt = counters[e];
  const int m0  = blockIdx.y * 32;
  if (m0 >= cnt) return;                 // uniform exit; EXEC all-1s at WMMA

  const int tid   = threadIdx.x;
  const int lane  = tid & 31;
  const int wave  = tid >> 5;
  const int ncol  = lane & 15;           // also A-fragment M index
  const int khalf = (lane >> 4) << 1;    // 0 for lanes 0-15, 2 for 16-31
  const int n0    = blockIdx.x * 128 + wave * 32;

  __shared__ __align__(16) float As[2][32][32];  // double-buffered [m][k] tile
  __shared__ const float* Arow[32];

  if (tid < 32) {
    const int c = m0 + tid;
    if (GATHER) {
      const int tok = (c < cnt) ? slot_token[e * CAP + c] : 0;
      Arow[tid] = Abase + (size_t)tok * KD;
    } else {
      Arow[tid] = Abase + (size_t)(e * CAP + c) * KD;
    }
  }
  __syncthreads();

  // LDS byte offsets of the two A buffers (low 32 bits of flat LDS address).
  const uint32_t As_off0 = (uint32_t)(size_t)&As[0][0][0];
  const uint32_t As_off1 = (uint32_t)(size_t)&As[1][0][0];

  const int r0 = tid >> 3;        // staging row 0..15 (and r0+16)
  const int kk = (tid & 7) << 2;  // staging k offset, 128b per load

  auto issue = [&](int k0, int buf) {
    const uint32_t base = buf ? As_off1 : As_off0;
    async_load_b128(base + (uint32_t)((r0 * 32 + kk) * 4),
                    Arow[r0] + k0 + kk);
    async_load_b128(base + (uint32_t)(((r0 + 16) * 32 + kk) * 4),
                    Arow[r0 + 16] + k0 + kk);
  };

  const float* Wb = Bbase + (size_t)e * KD * NTOT;
  v8f acc00 = {0.f, 0.f, 0.f, 0.f, 0.f, 0.f, 0.f, 0.f};
  v8f acc01 = acc00, acc10 = acc00, acc11 = acc00;

  constexpr int NCH = KD / 32;
  issue(0, 0);
  for (int ch = 0; ch < NCH; ++ch) {
    const int cur = ch & 1;
    wait_asynccnt0();     // my chunk-ch async loads have landed in LDS
    __syncthreads();      // everyone's have; idle buffer free for rewrite
    if (ch + 1 < NCH) issue((ch + 1) * 32, cur ^ 1);
    const int k0 = ch * 32;
#pragma unroll
    for (int ks = 0; ks < 32; ks += 4) {
      v2f a0, a1;
      a0.x = As[cur][ncol][ks + khalf];
      a0.y = As[cur][ncol][ks + khalf + 1];
      a1.x = As[cur][ncol + 16][ks + khalf];
      a1.y = As[cur][ncol + 16][ks + khalf + 1];
      const float* Bp = Wb + (size_t)(k0 + ks + khalf) * NTOT + n0 + ncol;
      v2f b0; b0.x = Bp[0];  b0.y = Bp[NTOT];
      v2f b1; b1.x = Bp[16]; b1.y = Bp[16 + NTOT];
      acc00 = wmma4(a0, b0, acc00);
      acc10 = wmma4(a1, b0, acc10);
      acc01 = wmma4(a0, b1, acc01);
      acc11 = wmma4(a1, b1, acc11);
    }
  }

  // C/D layout: VGPR j -> M = j (lanes 0-15) or j+8 (lanes 16-31), N = lane&15
  const int mb = m0 + ((lane < 16) ? 0 : 8);
#pragma unroll
  for (int j = 0; j < 8; ++j) {
    float v00 = acc00[j], v01 = acc01[j];
    float v10 = acc10[j], v11 = acc11[j];
    if (DO_SILU) {
      v00 = silu(v00); v01 = silu(v01);
      v10 = silu(v10); v11 = silu(v11);
    }
    float* op0 = Obuf + (size_t)(e * CAP + mb + j) * NTOT + n0 + ncol;
    op0[0]  = v00;
    op0[16] = v01;
    float* op1 = op0 + (size_t)16 * NTOT;
    op1[0]  = v10;
    op1[16] = v11;
  }
}

// out[t,h] = sum_k weight[t,k] * y[slot[t,k], h]  (gather -> deterministic)
__global__ __launch_bounds__(256)
void combine_kernel(const float* __restrict__ ybuf,
                    const int* __restrict__ pair_slot,
                    const float* __restrict__ pair_weight,
                    float* __restrict__ out) {
  const int t = blockIdx.x;
  for (int h = threadIdx.x; h < HDIM; h += 256) {
    float s = 0.f;
#pragma unroll
    for (int k = 0; k < TOPK; ++k) {
      const int sl = pair_slot[t * TOPK + k];
      if (sl >= 0)
        s = fmaf(pair_weight[t * TOPK + k], ybuf[(size_t)sl * HDIM + h], s);
    }
    out[(size_t)t * HDIM + h] = s;
  }
}

extern "C" void kernel_launch(void* const* d_in, const int* in_sizes, int n_in,
                              void* d_out, int out_size, void* d_ws,
                              size_t ws_size, hipStream_t stream) {
  (void)in_sizes; (void)n_in; (void)out_size; (void)ws_size;
  const float* x  = (const float*)d_in[0];  // [T,H]
  const float* gw = (const float*)d_in[1];  // [H,E]
  const float* w1 = (const float*)d_in[2];  // [E,H,I]
  const float* w2 = (const float*)d_in[3];  // [E,I,H]
  float* out = (float*)d_out;               // [T,H]

  // Workspace layout (256-B aligned offsets)
  char* ws = (char*)d_ws;
  int*   counters    = (int*)(ws + 0);                             // 64 ints
  int*   slot_token  = (int*)(ws + 256);                           // E*CAP ints
  int*   pair_slot   = (int*)(ws + 256 + 65536);                   // T*K ints
  float* pair_weight = (float*)(ws + 256 + 65536 + 24576);         // T*K floats
  float* hbuf        = (float*)(ws + 256 + 65536 + 24576 + 24576); // [E*CAP, I]
  float* ybuf        = (float*)((char*)hbuf +
                                (size_t)NEXP * CAP * IDIM * sizeof(float)); // [E*CAP, H]

  hipLaunchKernelGGL(init_counters, dim3(1), dim3(64), 0, stream, counters);
  hipLaunchKernelGGL(router_kernel, dim3(TTOK), dim3(64), 0, stream,
                     x, gw, counters, slot_token, pair_slot, pair_weight);
  hipLaunchKernelGGL((expert_gemm<HDIM, IDIM, true, true>),
                     dim3(IDIM / 128, CAP / 32, NEXP), dim3(128), 0, stream,
                     x, slot_token, w1, counters, hbuf);
  hipLaunchKernelGGL((expert_gemm<IDIM, HDIM, false, false>),
                     dim3(HDIM / 128, CAP / 32, NEXP), dim3(128), 0, stream,
                     hbuf, slot_token, w2, counters, ybuf);
  hipLaunchKernelGGL(combine_kernel, dim3(TTOK), dim3(256), 0, stream,
                     ybuf, pair_slot, pair_weight, out);
}